// DriftingModel_25099788878653
// MI455X (gfx1250) — compile-verified
//
#include <hip/hip_runtime.h>
#include <hip/hip_bf16.h>

typedef float v2f __attribute__((ext_vector_type(2)));
typedef float v8f __attribute__((ext_vector_type(8)));

#define N_ROWS 4096
#define N_COLS 8192  // pos (4096) + neg (4096)
#define DF 64
#define LDA 66       // LDS row stride: even (8B-aligned pairs) + conflict-free
#define TEMP_INV 20.0f
#define BIG_F 1000000.0f

// workspace float offsets
#define WS_ROWMAX 0
#define WS_COLMAX 4096
#define WS_ROWSUM 12288
#define WS_COLSUM 16384
#define WS_LSER   24576
#define WS_LSEC   28672
#define WS_SPOS   36864
#define WS_SNEG   40960
#define WS_XN     45056
#define WS_YN     49152
#define WS_TOTAL  57344

#if __has_builtin(__builtin_amdgcn_wmma_f32_16x16x4_f32)
#define USE_WMMA_F32X4 1
#else
#define USE_WMMA_F32X4 0
#endif

__device__ inline float atomicMaxF(float* addr, float val) {
  unsigned int* ua = (unsigned int*)addr;
  unsigned int old = __float_as_uint(*addr);
  while (__uint_as_float(old) < val) {
    unsigned int assumed = old;
    old = atomicCAS(ua, assumed, __float_as_uint(val));
    if (old == assumed) break;
  }
  return __uint_as_float(old);
}

// Dual-tile: C0 += A * B0^T, C1 += A * B1^T.
// A: 16m x 64k (stride LDA), B0/B1: 16n x 64k (stride LDA); all in LDS.
// C fragment layout (ISA 7.12.2): vgpr r, lanes 0-15: M=r, lanes 16-31: M=8+r; N=lane&15.
__device__ inline void mma_nt2(const float* A, const float* B0, const float* B1,
                               v8f& c0, v8f& c1, int lane) {
  const int mf = lane & 15;
  const int kh = (lane >> 4) << 1;  // 0 or 2
#if USE_WMMA_F32X4
#pragma unroll
  for (int k0 = 0; k0 < DF; k0 += 4) {
    const v2f a  = *(const v2f*)(A  + mf * LDA + k0 + kh);   // 8B aligned
    const v2f b0 = *(const v2f*)(B0 + mf * LDA + k0 + kh);
    const v2f b1 = *(const v2f*)(B1 + mf * LDA + k0 + kh);
    c0 = __builtin_amdgcn_wmma_f32_16x16x4_f32(false, a, false, b0, (short)0, c0, false, false);
    c1 = __builtin_amdgcn_wmma_f32_16x16x4_f32(false, a, false, b1, (short)0, c1, false, false);
  }
#else
  const int mrow = (lane >> 4) << 3;
#pragma unroll
  for (int r = 0; r < 8; ++r) {
    float a0 = c0[r], a1 = c1[r];
    const float* ar = A + (mrow + r) * LDA;
    const float* b0r = B0 + mf * LDA;
    const float* b1r = B1 + mf * LDA;
    for (int k = 0; k < DF; ++k) {
      a0 += ar[k] * b0r[k];
      a1 += ar[k] * b1r[k];
    }
    c0[r] = a0; c1[r] = a1;
  }
#endif
}

// Dual-tile: C0 += A * B[:, n0:n0+16], C1 += A * B[:, n1:n1+16].
// A: 16m x 64k (A[m][k], stride LDA); B: 64k x 64n (B[k][n], stride LDA).
__device__ inline void mma_nn2(const float* A, const float* B, int n0, int n1,
                               v8f& c0, v8f& c1, int lane) {
  const int nf = lane & 15;
  const int kh = (lane >> 4) << 1;
#if USE_WMMA_F32X4
#pragma unroll
  for (int k0 = 0; k0 < DF; k0 += 4) {
    const v2f a = *(const v2f*)(A + nf * LDA + k0 + kh);
    v2f b0, b1;
    b0[0] = B[(k0 + kh) * LDA + n0 + nf];
    b0[1] = B[(k0 + kh + 1) * LDA + n0 + nf];
    b1[0] = B[(k0 + kh) * LDA + n1 + nf];
    b1[1] = B[(k0 + kh + 1) * LDA + n1 + nf];
    c0 = __builtin_amdgcn_wmma_f32_16x16x4_f32(false, a, false, b0, (short)0, c0, false, false);
    c1 = __builtin_amdgcn_wmma_f32_16x16x4_f32(false, a, false, b1, (short)0, c1, false, false);
  }
#else
  const int mrow = (lane >> 4) << 3;
#pragma unroll
  for (int r = 0; r < 8; ++r) {
    float a0 = c0[r], a1 = c1[r];
    const float* ar = A + (mrow + r) * LDA;
    for (int k = 0; k < DF; ++k) {
      a0 += ar[k] * B[k * LDA + n0 + nf];
      a1 += ar[k] * B[k * LDA + n1 + nf];
    }
    c0[r] = a0; c1[r] = a1;
  }
#endif
}

// Zero out + initialize workspace reduction buffers.
__global__ __launch_bounds__(256) void init_kernel(float* __restrict__ ws,
                                                   float* __restrict__ out) {
  const int t = blockIdx.x * blockDim.x + threadIdx.x;
  if (t < N_ROWS * DF) out[t] = 0.0f;
  if (t < WS_TOTAL) ws[t] = (t < WS_ROWSUM) ? -3.0e38f : 0.0f;  // row/col max -> -inf
}

// Squared norms of x rows and (pos|neg) y rows.
__global__ __launch_bounds__(256) void norms_kernel(const float* __restrict__ x,
                                                    const float* __restrict__ ypos,
                                                    const float* __restrict__ yneg,
                                                    float* __restrict__ ws) {
  const int t = blockIdx.x * blockDim.x + threadIdx.x;
  if (t >= N_ROWS + N_COLS) return;
  const float* src;
  float* dst;
  if (t < N_ROWS) {
    src = x + t * DF;
    dst = ws + WS_XN + t;
  } else {
    const int j = t - N_ROWS;
    src = (j < N_ROWS) ? (ypos + j * DF) : (yneg + (j - N_ROWS) * DF);
    dst = ws + WS_YN + j;
  }
  float s = 0.0f;
  for (int k = 0; k < DF; ++k) s += src[k] * src[k];
  *dst = s;
}

__global__ __launch_bounds__(256) void lse_kernel(float* __restrict__ ws) {
  const int t = blockIdx.x * blockDim.x + threadIdx.x;
  if (t < N_ROWS) {
    ws[WS_LSER + t] = ws[WS_ROWMAX + t] + logf(ws[WS_ROWSUM + t]);
  } else if (t < N_ROWS + N_COLS) {
    const int j = t - N_ROWS;
    ws[WS_LSEC + j] = ws[WS_COLMAX + j] + logf(ws[WS_COLSUM + j]);
  }
}

// PHASE 1: row/col max of logits.  PHASE 2: row/col sum of exp(logit - max).
// PHASE 3: S_pos[i] = sum_j A_pos[i,j], S_neg[i] = sum_j A_neg[i,j].
template <int PHASE>
__global__ __launch_bounds__(256) void pass_kernel(const float* __restrict__ x,
                                                   const float* __restrict__ ypos,
                                                   const float* __restrict__ yneg,
                                                   float* __restrict__ ws) {
  float* rowMax = ws + WS_ROWMAX;
  float* colMax = ws + WS_COLMAX;
  float* rowSum = ws + WS_ROWSUM;
  float* colSum = ws + WS_COLSUM;
  float* Spos = ws + WS_SPOS;
  float* Sneg = ws + WS_SNEG;

  const int jt = blockIdx.x;  // 0..127 column tile (64 wide)
  const int it = blockIdx.y;  // 0..63 row tile
  const int i0 = it * 64;
  const int c0 = jt * 64;
  const bool isNeg = (jt >= 64);
  const float* y = isNeg ? yneg : ypos;
  const int j0 = isNeg ? (c0 - N_ROWS) : c0;

  __shared__ float XS[64][LDA];
  __shared__ float YS[64][LDA];
  __shared__ float XN[64], YN[64], RD[64], CD[64];

  const int tid = threadIdx.x;
  for (int i = tid; i < 1024; i += 256) {
    const int r = i >> 4;
    const int cc = (i & 15) << 2;
    const float4 xv = *reinterpret_cast<const float4*>(x + (i0 + r) * DF + cc);
    XS[r][cc] = xv.x; XS[r][cc + 1] = xv.y; XS[r][cc + 2] = xv.z; XS[r][cc + 3] = xv.w;
    const float4 yv = *reinterpret_cast<const float4*>(y + (j0 + r) * DF + cc);
    YS[r][cc] = yv.x; YS[r][cc + 1] = yv.y; YS[r][cc + 2] = yv.z; YS[r][cc + 3] = yv.w;
  }
  if (tid < 64) {
    XN[tid] = ws[WS_XN + i0 + tid];
  } else if (tid < 128) {
    YN[tid - 64] = ws[WS_YN + c0 + (tid - 64)];
  } else if (tid < 192) {
    const int t = tid - 128;
    if (PHASE == 2) RD[t] = ws[WS_ROWMAX + i0 + t];
    if (PHASE == 3) RD[t] = ws[WS_LSER + i0 + t];
  } else {
    const int t = tid - 192;
    if (PHASE == 2) CD[t] = ws[WS_COLMAX + c0 + t];
    if (PHASE == 3) CD[t] = ws[WS_LSEC + c0 + t];
  }
  __syncthreads();

  const int wv = tid >> 5;
  const int lane = tid & 31;
  const int mf = lane & 15;
  const int mrow = (lane >> 4) << 3;

  // Wave wv owns subtiles s = 2wv, 2wv+1: same ti, adjacent tj => shared A.
  const int ti = wv >> 1;
  const int tj0 = (wv & 1) << 1;  // 0 or 2

  v8f cc0 = {0.f, 0.f, 0.f, 0.f, 0.f, 0.f, 0.f, 0.f};
  v8f cc1 = cc0;
  mma_nt2(&XS[ti * 16][0], &YS[tj0 * 16][0], &YS[(tj0 + 1) * 16][0], cc0, cc1, lane);

#pragma unroll
  for (int q = 0; q < 2; ++q) {
    const v8f c = q ? cc1 : cc0;
    const int tj = tj0 + q;
    const int nj = tj * 16 + mf;
    const int gc = c0 + nj;
    float colacc = (PHASE == 1) ? -3.0e38f : 0.0f;
#pragma unroll
    for (int r = 0; r < 8; ++r) {
      const int mi = ti * 16 + mrow + r;
      const int gi = i0 + mi;
      const float sq = XN[mi] + YN[nj] - 2.0f * c[r];
      float d = sqrtf(fmaxf(sq, 1e-12f));
      if (isNeg && gi == gc - N_ROWS) d += BIG_F;  // eye * BIG on dist_neg
      const float logit = -d * TEMP_INV;

      if (PHASE == 1) {
        float rv = logit;
#pragma unroll
        for (int off = 1; off < 16; off <<= 1) rv = fmaxf(rv, __shfl_xor(rv, off, 32));
        if (mf == 0) atomicMaxF(rowMax + gi, rv);
        colacc = fmaxf(colacc, logit);
      } else if (PHASE == 2) {
        float rs = expf(logit - RD[mi]);
#pragma unroll
        for (int off = 1; off < 16; off <<= 1) rs += __shfl_xor(rs, off, 32);
        if (mf == 0) atomicAdd(rowSum + gi, rs);
        colacc += expf(logit - CD[nj]);
      } else {  // PHASE 3
        float rs = expf(logit - 0.5f * (RD[mi] + CD[nj]));
#pragma unroll
        for (int off = 1; off < 16; off <<= 1) rs += __shfl_xor(rs, off, 32);
        if (mf == 0) atomicAdd((isNeg ? Sneg : Spos) + gi, rs);
      }
    }
    if (PHASE == 1) {
      colacc = fmaxf(colacc, __shfl_xor(colacc, 16, 32));
      if (lane < 16) atomicMaxF(colMax + gc, colacc);
    } else if (PHASE == 2) {
      colacc += __shfl_xor(colacc, 16, 32);
      if (lane < 16) atomicAdd(colSum + gc, colacc);
    }
  }
}

// PHASE 4: recompute A, scale into W, and do out = W_pos@y_pos - W_neg@y_neg
// with a second WMMA chain; partial results atomically added into d_out.
__global__ __launch_bounds__(256) void pass4_kernel(const float* __restrict__ x,
                                                    const float* __restrict__ ypos,
                                                    const float* __restrict__ yneg,
                                                    const float* __restrict__ ws,
                                                    float* __restrict__ out) {
  const int slice = blockIdx.x;  // 0..15, each covers 8 column tiles (512 cols)
  const int it = blockIdx.y;     // 0..63
  const int i0 = it * 64;

  __shared__ float XS[64][LDA];
  __shared__ float YS[64][LDA];
  __shared__ float WSH[64][LDA];
  __shared__ float XN[64], LR[64], SP[64], SN[64], YN[64], LC[64];

  const int tid = threadIdx.x;
  for (int i = tid; i < 1024; i += 256) {
    const int r = i >> 4;
    const int cc = (i & 15) << 2;
    const float4 xv = *reinterpret_cast<const float4*>(x + (i0 + r) * DF + cc);
    XS[r][cc] = xv.x; XS[r][cc + 1] = xv.y; XS[r][cc + 2] = xv.z; XS[r][cc + 3] = xv.w;
  }
  if (tid < 64) XN[tid] = ws[WS_XN + i0 + tid];
  else if (tid < 128) LR[tid - 64] = ws[WS_LSER + i0 + (tid - 64)];
  else if (tid < 192) SP[tid - 128] = ws[WS_SPOS + i0 + (tid - 128)];
  else SN[tid - 192] = ws[WS_SNEG + i0 + (tid - 192)];
  __syncthreads();

  const int wv = tid >> 5;
  const int lane = tid & 31;
  const int mf = lane & 15;
  const int mrow = (lane >> 4) << 3;
  const int ti = wv >> 1;
  const int tj0 = (wv & 1) << 1;

  v8f acc0 = {0.f, 0.f, 0.f, 0.f, 0.f, 0.f, 0.f, 0.f};
  v8f acc1 = acc0;

  for (int ch = 0; ch < 8; ++ch) {
    const int jt = slice * 8 + ch;
    const bool isNeg = (jt >= 64);
    const float* y = isNeg ? yneg : ypos;
    const int c0 = jt * 64;
    const int j0 = isNeg ? (c0 - N_ROWS) : c0;

    // Prefetch next chunk's y tile while this one is being consumed.
    if (ch + 1 < 8 && tid < 128) {
      const int jt2 = jt + 1;
      const float* y2 = (jt2 >= 64) ? yneg : ypos;
      const int j2 = (jt2 >= 64) ? (jt2 * 64 - N_ROWS) : (jt2 * 64);
      const int rr = tid >> 1;
      const int oo = (tid & 1) << 5;
      __builtin_prefetch(y2 + (j2 + rr) * DF + oo, 0, 1);
    }

    __syncthreads();  // previous iteration done with YS/WSH
    for (int i = tid; i < 1024; i += 256) {
      const int r = i >> 4;
      const int cc = (i & 15) << 2;
      const float4 yv = *reinterpret_cast<const float4*>(y + (j0 + r) * DF + cc);
      YS[r][cc] = yv.x; YS[r][cc + 1] = yv.y; YS[r][cc + 2] = yv.z; YS[r][cc + 3] = yv.w;
    }
    if (tid < 64) YN[tid] = ws[WS_YN + c0 + tid];
    else if (tid < 128) LC[tid - 64] = ws[WS_LSEC + c0 + (tid - 64)];
    __syncthreads();

    // logits -> A -> W tile into LDS
    {
      v8f cc0 = {0.f, 0.f, 0.f, 0.f, 0.f, 0.f, 0.f, 0.f};
      v8f cc1 = cc0;
      mma_nt2(&XS[ti * 16][0], &YS[tj0 * 16][0], &YS[(tj0 + 1) * 16][0], cc0, cc1, lane);
#pragma unroll
      for (int q = 0; q < 2; ++q) {
        const v8f c = q ? cc1 : cc0;
        const int tj = tj0 + q;
        const int nj = tj * 16 + mf;
        const int gc = c0 + nj;
#pragma unroll
        for (int r = 0; r < 8; ++r) {
          const int mi = ti * 16 + mrow + r;
          const int gi = i0 + mi;
          const float sq = XN[mi] + YN[nj] - 2.0f * c[r];
          float d = sqrtf(fmaxf(sq, 1e-12f));
          if (isNeg && gi == gc - N_ROWS) d += BIG_F;
          const float logit = -d * TEMP_INV;
          const float a = expf(logit - 0.5f * (LR[mi] + LC[nj]));
          WSH[mi][nj] = isNeg ? -(a * SP[mi]) : (a * SN[mi]);
        }
      }
    }
    __syncthreads();

    // out(64m x 64d) += W(64m x 64n) @ Y(64n x 64d); shared A = W row block.
    mma_nn2(&WSH[ti * 16][0], &YS[0][0], tj0 * 16, (tj0 + 1) * 16, acc0, acc1, lane);
  }

#pragma unroll
  for (int q = 0; q < 2; ++q) {
    const int tjo = tj0 + q;
    const v8f a = q ? acc1 : acc0;
#pragma unroll
    for (int r = 0; r < 8; ++r) {
      const int gi = i0 + ti * 16 + mrow + r;
      const int dcol = tjo * 16 + mf;
      atomicAdd(out + gi * DF + dcol, a[r]);
    }
  }
}

extern "C" void kernel_launch(void* const* d_in, const int* in_sizes, int n_in,
                              void* d_out, int out_size, void* d_ws, size_t ws_size,
                              hipStream_t stream) {
  (void)in_sizes; (void)n_in; (void)out_size; (void)ws_size;
  const float* x = (const float*)d_in[0];
  const float* ypos = (const float*)d_in[1];
  const float* yneg = (const float*)d_in[2];
  float* out = (float*)d_out;
  float* ws = (float*)d_ws;

  init_kernel<<<dim3((N_ROWS * DF + 255) / 256), 256, 0, stream>>>(ws, out);
  norms_kernel<<<dim3((N_ROWS + N_COLS + 255) / 256), 256, 0, stream>>>(x, ypos, yneg, ws);

  const dim3 tg(128, 64);
  pass_kernel<1><<<tg, 256, 0, stream>>>(x, ypos, yneg, ws);
  pass_kernel<2><<<tg, 256, 0, stream>>>(x, ypos, yneg, ws);
  lse_kernel<<<dim3((N_ROWS + N_COLS + 255) / 256), 256, 0, stream>>>(ws);
  pass_kernel<3><<<tg, 256, 0, stream>>>(x, ypos, yneg, ws);
  pass4_kernel<<<dim3(16, 64), 256, 0, stream>>>(x, ypos, yneg, ws, out);
}